// NeighborLayer2D_44452911513681
// MI455X (gfx1250) — compile-verified
//
#include <hip/hip_runtime.h>

typedef __attribute__((ext_vector_type(2))) float v2f;
typedef __attribute__((ext_vector_type(8))) float v8f;
typedef __attribute__((ext_vector_type(4))) unsigned int v4u;
typedef __attribute__((ext_vector_type(4))) int v4i;
typedef __attribute__((ext_vector_type(8))) int v8i;

#define BATCH 4
#define CIN   32
#define COUT  32
#define HH    68
#define WW    68
#define KH    5
#define KW    5
#define KTOT  800            // CIN*KH*KW
#define NPOS  4096           // OH*OW
#define RSTR  850            // padded LDS row stride in dwords (800 + 2 per 32)
#define WLDW  (RSTR * COUT)  // 27200 dwords for W^T slice
#define PLDW  (RSTR * BATCH) // 3400 dwords for patches

// TDM LDS pad: +2 dwords every 32 stored dwords -> lds dword(i) = i + 2*(i>>5)
__device__ __forceinline__ int padoff(int k) { return k + ((k >> 5) << 1); }

// One block (256 threads = 8 waves) per output position (y,x).
// TDM streams the 100KB weight slice + 12.8KB patch into LDS, then
// C(16x16 f32) = W^T (Cout x K) * patch^T (K x B) via v_wmma_f32_16x16x4_f32.
__global__ __launch_bounds__(256)
void lc2d_wmma_kernel(const float* __restrict__ x,
                      const float* __restrict__ weight,
                      const float* __restrict__ bias,
                      float* __restrict__ out) {
  extern __shared__ float smem[];
  // layout (dwords): [0, WLDW)       Wl : row d (stride RSTR), elem k at padoff(k)
  //                  [WLDW, +PLDW)   Pl : row b (stride RSTR), elem k at padoff(k)
  //                  [+512]          Cs : partial-C scratch
  float* Pl = smem + WLDW;
  float* Cs = Pl + PLDW;

  const int pos = blockIdx.x;          // pos = y*64 + x
  const int y0  = pos >> 6;
  const int x0  = pos & 63;
  const int tid = threadIdx.x;

#if __has_builtin(__builtin_amdgcn_tensor_load_to_lds)
  if (tid == 0) {
    const unsigned int ldsW = (unsigned int)(uintptr_t)smem;
    const unsigned int ldsP = (unsigned int)(uintptr_t)Pl;
    // flags: data_size=4B, pad_enable, pad_interval=32dw (code 4), pad_amount=2dw (code 1)
    const unsigned int flags = (2u << 16) | (1u << 20) | (4u << 22) | (1u << 25);

    // ---- weight slice: x=j[25] (contig), y=c[32] stride 32*102400, z=d[32] stride 102400
    // LDS order (z outer) = [d][c][j]  ==  Wl[d][k=c*25+j]
    {
      unsigned long long ga =
          (unsigned long long)(uintptr_t)weight + (unsigned long long)pos * 100u;
      v4u g0 = {1u, ldsW, (unsigned int)ga,
                (unsigned int)(ga >> 32) | (2u << 30)};
      v8i g1 = {(int)flags,
                (int)(25u << 16),                 // tensor_dim0 = 25
                (int)(32u << 16),                 // tensor_dim1 = 32
                (int)(25u << 16),                 // tile_dim0 = 25
                (int)(32u | (32u << 16)),         // tile_dim1 = 32, tile_dim2 = 32
                (int)3276800u,                    // tensor_dim0_stride (c-stride)
                (int)((102400u & 0xffffu) << 16), // tensor_dim1_stride lo (d-stride)
                (int)(102400u >> 16)};            // tensor_dim1_stride hi
      v4i g2 = {32, 0, 0, 0};                     // tensor_dim2 = 32
      v4i g3 = {0, 0, 0, 0};
#if defined(__clang_major__) && __clang_major__ >= 23
      v8i gz = {0, 0, 0, 0, 0, 0, 0, 0};
      __builtin_amdgcn_tensor_load_to_lds(g0, g1, g2, g3, gz, 0);
#else
      __builtin_amdgcn_tensor_load_to_lds(g0, g1, g2, g3, 0);
#endif
    }
    // ---- patch: x=w[5], y=h[5] s=68, z=c[32] s=4624, zz=b[4] s=147968
    // LDS order = [b][c][h][w] == Pl[b][k=c*25+h*5+w]
    {
      unsigned long long ga = (unsigned long long)(uintptr_t)x +
                              (unsigned long long)(y0 * WW + x0) * 4u;
      v4u g0 = {1u, ldsP, (unsigned int)ga,
                (unsigned int)(ga >> 32) | (2u << 30)};
      v8i g1 = {(int)flags,
                (int)(5u << 16),                  // tensor_dim0 = 5
                (int)(5u << 16),                  // tensor_dim1 = 5
                (int)(5u << 16),                  // tile_dim0 = 5
                (int)(5u | (32u << 16)),          // tile_dim1 = 5, tile_dim2 = 32
                (int)68u,                         // tensor_dim0_stride (h-stride)
                (int)((4624u & 0xffffu) << 16),   // tensor_dim1_stride lo (c-stride)
                (int)(4624u >> 16)};
      v4i g2 = {32, 4, 147968, (int)(4u << 16)};  // dim2=32, dim3=4, dim2_stride, tile_dim3=4
      v4i g3 = {0, 0, 0, 0};
#if defined(__clang_major__) && __clang_major__ >= 23
      v8i gz = {0, 0, 0, 0, 0, 0, 0, 0};
      __builtin_amdgcn_tensor_load_to_lds(g0, g1, g2, g3, gz, 0);
#else
      __builtin_amdgcn_tensor_load_to_lds(g0, g1, g2, g3, 0);
#endif
    }
    __builtin_amdgcn_s_wait_tensorcnt(0);
  }
#else
  // ---- fallback: manual staging (same padded layout) ----
  for (int e = tid; e < CIN * COUT * KH * KW; e += 256) {
    int cd = e / 25, j = e - cd * 25;
    int c = cd >> 5, d = cd & 31;
    smem[d * RSTR + padoff(c * 25 + j)] =
        weight[(size_t)cd * (NPOS * 25) + (size_t)pos * 25 + j];
  }
  for (int e = tid; e < BATCH * KTOT; e += 256) {
    int b = e / KTOT, k = e - b * KTOT;
    int c = k / 25, r = k - c * 25, h = r / 5, w = r - h * 5;
    Pl[b * RSTR + padoff(k)] =
        x[((b * CIN + c) * HH + (y0 + h)) * WW + (x0 + w)];
  }
#endif
  __syncthreads();

  const int wave  = tid >> 5;
  const int lane  = tid & 31;
  const int mi    = lane & 15;   // M row (A) / N col (B,C)
  const int hi    = lane >> 4;   // half-wave select
  const int tile  = wave & 1;    // Cout tile: d in [tile*16, tile*16+16)
  const int khalf = (wave >> 1) & 1;
  v8f acc = {};

  if (wave < 4) {
    // A fragment (16x4 f32): lane m = mi holds K = k0+2*hi, +1 ; A[m][k] = W(k, tile*16+m)
    const float* Abase = smem + (tile * 16 + mi) * RSTR;
    // B fragment (4x16 f32): lane n = mi holds K = k0+2*hi, +1 ; B[k][n] = patch(b=n&3, k)
    const float* Bbase = Pl + (mi & 3) * RSTR;

    const int kbeg = khalf * 400;
#pragma unroll 4
    for (int k0 = kbeg; k0 < kbeg + 400; k0 += 4) {
      int ka = k0 + 2 * hi;
      int t  = ka + ((ka >> 5) << 1);   // padded offset, stays even & pair-contiguous
      v2f a = *(const v2f*)(Abase + t);
      v2f b = *(const v2f*)(Bbase + t);
      acc = __builtin_amdgcn_wmma_f32_16x16x4_f32(
          /*neg_a=*/false, a, /*neg_b=*/false, b,
          /*c_mod=*/(short)0, acc, /*reuse_a=*/false, /*reuse_b=*/false);
    }
    if (khalf == 1) {
#pragma unroll
      for (int r = 0; r < 8; ++r) Cs[tile * 256 + r * 32 + lane] = acc[r];
    }
  }
  __syncthreads();

  if (wave < 2) {
    // wave w holds K-half-0 partial for tile w; add K-half-1 partial + bias.
    // C layout: lane n = mi; VGPR r holds M = r (lanes 0-15) / r+8 (lanes 16-31)
#pragma unroll
    for (int r = 0; r < 8; ++r) {
      int m = r + 8 * hi;
      int d = tile * 16 + m;
      float val = acc[r] + Cs[tile * 256 + r * 32 + lane]
                + bias[d * NPOS + pos];
      if (mi < BATCH)
        out[((size_t)mi * COUT + d) * NPOS + pos] = val;
    }
  }
}

extern "C" void kernel_launch(void* const* d_in, const int* in_sizes, int n_in,
                              void* d_out, int out_size, void* d_ws, size_t ws_size,
                              hipStream_t stream) {
  const float* x      = (const float*)d_in[0];
  const float* weight = (const float*)d_in[1];
  const float* bias   = (const float*)d_in[2];
  float* out          = (float*)d_out;

  const size_t smem_bytes = (size_t)(WLDW + PLDW + 2 * 256) * sizeof(float);
  (void)hipFuncSetAttribute((const void*)lc2d_wmma_kernel,
                            hipFuncAttributeMaxDynamicSharedMemorySize,
                            (int)smem_bytes);
  lc2d_wmma_kernel<<<NPOS, 256, smem_bytes, stream>>>(x, weight, bias, out);
}